// MMCL_inv_2138893713973
// MI455X (gfx1250) — compile-verified
//
#include <hip/hip_runtime.h>
#include <hip/hip_bf16.h>

typedef float v2f __attribute__((ext_vector_type(2)));
typedef float v8f __attribute__((ext_vector_type(8)));

#define BS 384          // batch size / matrix dim
#define NF 768          // 2*BS columns of K
#define DK 128          // feature dim
#define GAMMA 14.285714285714286f   // 1/0.07
#define NS_ITERS 30     // Newton-Schulz iterations (cond(B)~350 -> ~22 needed)

#define WMMA_F32(A, B, C) \
    __builtin_amdgcn_wmma_f32_16x16x4_f32(false, (A), false, (B), (short)0, (C), false, false)

// ---------------------------------------------------------------------------
// WMMA f32 16x16x4 lane mapping (wave32):
//   lo = lane & 15, hi = lane >> 4
//   A (16x4):  reg r -> A[m = lo][k = 2*hi + r]
//   B (4x16):  reg r -> B[k = 2*hi + r][n = lo]
//   C/D (16x16): reg v -> C[m = v + 8*hi][n = lo]
// Each wave computes a 32x32 macro-tile (2x2 accumulators) so every k-step
// issues 4 WMMAs against 4 fragment loads (1:1), with 4 independent
// accumulation chains to hide WMMA latency.
// ---------------------------------------------------------------------------

__device__ __forceinline__ void mmcl_store_tile_kmat(
    float* __restrict__ Km, float* __restrict__ Bm, float* __restrict__ KKd0,
    int m0, int n, int hi, v8f acc)
{
    #pragma unroll
    for (int v = 0; v < 8; ++v) {
        const int m = m0 + v + 8 * hi;
        const float kv = __expf(GAMMA * (2.0f * acc[v] - 2.0f));
        Km[(size_t)m * NF + n] = kv;
        if (n < BS) {
            const float d = (m == n) ? 1.0f : 0.0f;
            Bm[(size_t)m * BS + n]   = 1.0f + 0.1f * d + kv;
            KKd0[(size_t)m * BS + n] = (m == n) ? 0.0f : kv;
        }
    }
}

// K[a,c] = exp(gamma*(2*F0a.Fc - 2)); also B = 1 + 0.1*eye + KK, KKd0 = KK w/ zero diag
__global__ __launch_bounds__(128) void mmcl_kmat_kernel(
    const float* __restrict__ feats, float* __restrict__ Km,
    float* __restrict__ Bm, float* __restrict__ KKd0)
{
    const int lane = threadIdx.x & 31;
    const int wave = threadIdx.x >> 5;
    const int lo = lane & 15, hi = lane >> 4;
    const int row0 = blockIdx.y * 64 + (wave >> 1) * 32;   // 32-row macro-tile
    const int col0 = blockIdx.x * 64 + (wave & 1) * 32;    // 32-col macro-tile
    const int n0 = col0 + lo, n1 = col0 + 16 + lo;

    // A: view-0 feature rows; feats layout (384,2,128)
    const float* arow0 = feats + (size_t)(row0 + lo) * 256;
    const float* arow1 = feats + (size_t)(row0 + 16 + lo) * 256;
    // B: column n of F^T == feature row n (view 0 for n<384, view 1 otherwise)
    const float* brow0 = (n0 < BS) ? (feats + (size_t)n0 * 256)
                                   : (feats + (size_t)(n0 - BS) * 256 + 128);
    const float* brow1 = (n1 < BS) ? (feats + (size_t)n1 * 256)
                                   : (feats + (size_t)(n1 - BS) * 256 + 128);

    v8f acc00 = {}, acc01 = {}, acc10 = {}, acc11 = {};
    #pragma unroll 2
    for (int k0 = 0; k0 < DK; k0 += 4) {
        const int ka = k0 + 2 * hi;
        v2f a0, a1, b0, b1;
        a0.x = arow0[ka]; a0.y = arow0[ka + 1];
        a1.x = arow1[ka]; a1.y = arow1[ka + 1];
        b0.x = brow0[ka]; b0.y = brow0[ka + 1];
        b1.x = brow1[ka]; b1.y = brow1[ka + 1];
        acc00 = WMMA_F32(a0, b0, acc00);
        acc01 = WMMA_F32(a0, b1, acc01);
        acc10 = WMMA_F32(a1, b0, acc10);
        acc11 = WMMA_F32(a1, b1, acc11);
    }
    mmcl_store_tile_kmat(Km, Bm, KKd0, row0,      n0, hi, acc00);
    mmcl_store_tile_kmat(Km, Bm, KKd0, row0,      n1, hi, acc01);
    mmcl_store_tile_kmat(Km, Bm, KKd0, row0 + 16, n0, hi, acc10);
    mmcl_store_tile_kmat(Km, Bm, KKd0, row0 + 16, n1, hi, acc11);
}

__device__ __forceinline__ void mmcl_store_tile_gemm(
    float* __restrict__ C, const float* __restrict__ D, float alpha, float beta,
    int N, int m0, int n, int hi, v8f acc)
{
    #pragma unroll
    for (int v = 0; v < 8; ++v) {
        const int m = m0 + v + 8 * hi;
        const size_t off = (size_t)m * N + n;
        float r = alpha * acc[v];
        if (beta != 0.0f) r += beta * D[off];
        C[off] = r;
    }
}

// C = alpha*A@B + beta*D   (M=384 fixed, row-major, lda=K, ldb=ldc=ldd=N)
__global__ __launch_bounds__(128) void mmcl_gemm_kernel(
    float* __restrict__ C, const float* __restrict__ A, const float* __restrict__ Bm,
    const float* __restrict__ D, float alpha, float beta, int N, int K)
{
    const int lane = threadIdx.x & 31;
    const int wave = threadIdx.x >> 5;
    const int lo = lane & 15, hi = lane >> 4;
    const int row0 = blockIdx.y * 64 + (wave >> 1) * 32;
    const int col0 = blockIdx.x * 64 + (wave & 1) * 32;
    const int n0 = col0 + lo, n1 = col0 + 16 + lo;

    const float* arow0 = A + (size_t)(row0 + lo) * K;
    const float* arow1 = A + (size_t)(row0 + 16 + lo) * K;
    const float* bcol0 = Bm + n0;
    const float* bcol1 = Bm + n1;

    v8f acc00 = {}, acc01 = {}, acc10 = {}, acc11 = {};
    #pragma unroll 2
    for (int k0 = 0; k0 < K; k0 += 4) {
        const int ka = k0 + 2 * hi;
        v2f a0, a1, b0, b1;
        a0.x = arow0[ka];                 a0.y = arow0[ka + 1];
        a1.x = arow1[ka];                 a1.y = arow1[ka + 1];
        b0.x = bcol0[(size_t)ka * N];     b0.y = bcol0[(size_t)(ka + 1) * N];
        b1.x = bcol1[(size_t)ka * N];     b1.y = bcol1[(size_t)(ka + 1) * N];
        acc00 = WMMA_F32(a0, b0, acc00);
        acc01 = WMMA_F32(a0, b1, acc01);
        acc10 = WMMA_F32(a1, b0, acc10);
        acc11 = WMMA_F32(a1, b1, acc11);
    }
    mmcl_store_tile_gemm(C, D, alpha, beta, N, row0,      n0, hi, acc00);
    mmcl_store_tile_gemm(C, D, alpha, beta, N, row0,      n1, hi, acc01);
    mmcl_store_tile_gemm(C, D, alpha, beta, N, row0 + 16, n0, hi, acc10);
    mmcl_store_tile_gemm(C, D, alpha, beta, N, row0 + 16, n1, hi, acc11);
}

// per-row (abs-)sum of a 384x384 matrix
__global__ __launch_bounds__(128) void mmcl_rowsum_kernel(
    const float* __restrict__ M, float* __restrict__ outv, int absmode)
{
    __shared__ float red[128];
    const int r = blockIdx.x, t = threadIdx.x;
    float s = 0.0f;
    for (int c = t; c < BS; c += 128) {
        const float v = M[(size_t)r * BS + c];
        s += absmode ? fabsf(v) : v;
    }
    red[t] = s; __syncthreads();
    #pragma unroll
    for (int st = 64; st > 0; st >>= 1) {
        if (t < st) red[t] += red[t + st];
        __syncthreads();
    }
    if (t == 0) outv[r] = red[0];
}

// mode 0: out[0] = 1/(max^2);  mode 1: out[0] = sum
__global__ __launch_bounds__(384) void mmcl_reduce384_kernel(
    const float* __restrict__ inv, float* __restrict__ out, int mode)
{
    __shared__ float red[384];
    const int t = threadIdx.x;
    red[t] = inv[t]; __syncthreads();
    if (mode == 0) {
        if (t < 128) red[t] = fmaxf(red[t], fmaxf(red[t + 128], red[t + 256]));
        __syncthreads();
        #pragma unroll
        for (int s = 64; s > 0; s >>= 1) {
            if (t < s) red[t] = fmaxf(red[t], red[t + s]);
            __syncthreads();
        }
        if (t == 0) out[0] = 1.0f / (red[0] * red[0]);
    } else {
        if (t < 128) red[t] += red[t + 128] + red[t + 256];
        __syncthreads();
        #pragma unroll
        for (int s = 64; s > 0; s >>= 1) {
            if (t < s) red[t] += red[t + s];
            __syncthreads();
        }
        if (t == 0) out[0] = red[0];
    }
}

// X0 = B * scal[0]
__global__ __launch_bounds__(256) void mmcl_scale_kernel(
    float* __restrict__ X, const float* __restrict__ B, const float* __restrict__ scal)
{
    const int idx = blockIdx.x * 256 + threadIdx.x;
    if (idx < BS * BS) X[idx] = B[idx] * scal[0];
}

__device__ __forceinline__ float blockReduceSum384(float v, float* red)
{
    const int t = threadIdx.x;
    red[t] = v; __syncthreads();
    if (t < 128) red[t] += red[t + 128] + red[t + 256];
    __syncthreads();
    #pragma unroll
    for (int s = 64; s > 0; s >>= 1) {
        if (t < s) red[t] += red[t + s];
        __syncthreads();
    }
    const float r = red[0];
    __syncthreads();
    return r;
}

// Per-anchor Woodbury + bordered-deletion solve and loss contributions.
// one block (384 threads) per i; thread j handles row j.
__global__ __launch_bounds__(384) void mmcl_solve_kernel(
    const float* __restrict__ X, const float* __restrict__ G,
    const float* __restrict__ KKd0, const float* __restrict__ Km,
    const float* __restrict__ p, const float* __restrict__ scal,
    float* __restrict__ pos_part, float* __restrict__ neg_part)
{
    __shared__ float red[384];
    __shared__ float sh[2];
    const int i = blockIdx.x;
    const int j = threadIdx.x;

    const float s_pp = scal[1];                 // 1^T X 1
    const float Xji  = X[(size_t)j * BS + i];   // column i of X (symmetric)
    const float Gji  = G[(size_t)j * BS + i];   // (X k_i)[j]
    const float pj   = p[j];                    // (X 1)[j]
    const float p_i  = p[i];
    const float g_ii = G[(size_t)i * BS + i];   // k_i^T X e_i

    const float a_i = blockReduceSum384(Gji, red);                              // 1^T X k_i
    const float c_i = blockReduceSum384(KKd0[(size_t)j * BS + i] * Gji, red);   // k_i^T X k_i

    // M = [[a-1, s_pp],[c, a-1]], Woodbury capacitance (C = -I2)
    const float am1 = a_i - 1.0f;
    const float dinv = 1.0f / (am1 * am1 - s_pp * c_i);
    // rhs0 = 2(1 - e_i):  V^T X rhs0 = [2(s_pp - p_i); 2(a_i - g_ii)]
    const float v00 = 2.0f * (s_pp - p_i), v01 = 2.0f * (a_i - g_ii);
    const float w00 = dinv * ( am1 * v00 - s_pp * v01);
    const float w01 = dinv * (-c_i * v00 + am1 * v01);
    // rhs1 = e_i:  V^T X rhs1 = [p_i; g_ii]
    const float w10 = dinv * ( am1 * p_i - s_pp * g_ii);
    const float w11 = dinv * (-c_i * p_i + am1 * g_ii);

    const float x0 = 2.0f * pj - 2.0f * Xji - (Gji * w00 + pj * w01);
    const float x1 = Xji - (Gji * w10 + pj * w11);

    if (j == i) { sh[0] = x0; sh[1] = x1; }
    __syncthreads();
    const float x = x0 - (sh[0] / sh[1]) * x1;   // enforces x[i] == 0

    float alpha = fminf(fmaxf(x, 0.0f), 1.0f);   // relu + clip to C=1
    if (j == i) alpha = 0.0f;
    __syncthreads();

    const float ax   = blockReduceSum384(alpha, red);
    const float Ksji = Km[(size_t)j * NF + BS + i];          // Ks[j, i]
    const float ng   = blockReduceSum384(alpha * Ksji, red);

    if (j == 0) {
        const float Ksii = Km[(size_t)i * NF + BS + i];      // diag(Ks)[i]
        pos_part[i] = ax * Ksii;
        neg_part[i] = ng;
    }
}

__global__ __launch_bounds__(384) void mmcl_final_kernel(
    const float* __restrict__ pos_part, const float* __restrict__ neg_part,
    float* __restrict__ out)
{
    __shared__ float red[384];
    const int t = threadIdx.x;
    const float ps = blockReduceSum384(pos_part[t], red);
    const float ns = blockReduceSum384(neg_part[t], red);
    if (t == 0) {
        out[0] = -(ps / (float)BS);
        out[1] =  ns / (float)BS;
    }
}

extern "C" void kernel_launch(void* const* d_in, const int* in_sizes, int n_in,
                              void* d_out, int out_size, void* d_ws, size_t ws_size,
                              hipStream_t stream)
{
    const float* feats = (const float*)d_in[0];
    float* out = (float*)d_out;
    float* ws  = (float*)d_ws;

    float* Km   = ws;                        // 384*768
    float* Bm   = Km   + (size_t)BS * NF;    // 384*384
    float* KKd0 = Bm   + (size_t)BS * BS;
    float* Xa   = KKd0 + (size_t)BS * BS;
    float* Xb   = Xa   + (size_t)BS * BS;
    float* T    = Xb   + (size_t)BS * BS;
    float* G    = T    + (size_t)BS * BS;
    float* p    = G    + (size_t)BS * BS;
    float* tmpv = p + BS;
    float* pos_part = tmpv + BS;
    float* neg_part = pos_part + BS;
    float* scal     = neg_part + BS;
    const size_t need = (size_t)(scal + 8 - ws) * sizeof(float);
    if (ws_size < need) return;              // ~4.8 MB scratch required

    // 1) K = exp(gamma*(2*F0 F^T - 2)); B; KK (zero-diag)      [WMMA]
    mmcl_kmat_kernel<<<dim3(NF / 64, BS / 64), 128, 0, stream>>>(feats, Km, Bm, KKd0);

    // 2) Newton-Schulz init: X0 = B / (||B||inf)^2 (B symmetric)
    mmcl_rowsum_kernel<<<dim3(BS), 128, 0, stream>>>(Bm, tmpv, 1);
    mmcl_reduce384_kernel<<<1, 384, 0, stream>>>(tmpv, &scal[0], 0);
    mmcl_scale_kernel<<<dim3((BS * BS + 255) / 256), 256, 0, stream>>>(Xa, Bm, scal);

    // 3) X <- 2X - X B X  (all 384^3 WMMA GEMMs)
    float* X = Xa; float* X2 = Xb;
    for (int it = 0; it < NS_ITERS; ++it) {
        mmcl_gemm_kernel<<<dim3(BS / 64, BS / 64), 128, 0, stream>>>(
            T, Bm, X, nullptr, 1.0f, 0.0f, BS, BS);
        mmcl_gemm_kernel<<<dim3(BS / 64, BS / 64), 128, 0, stream>>>(
            X2, X, T, X, -1.0f, 2.0f, BS, BS);
        float* sw = X; X = X2; X2 = sw;
    }

    // 4) G = X @ KKd0  (columns are X k_i for every anchor)    [WMMA]
    mmcl_gemm_kernel<<<dim3(BS / 64, BS / 64), 128, 0, stream>>>(
        G, X, KKd0, nullptr, 1.0f, 0.0f, BS, BS);

    // 5) p = X 1, s_pp = 1^T X 1
    mmcl_rowsum_kernel<<<dim3(BS), 128, 0, stream>>>(X, p, 0);
    mmcl_reduce384_kernel<<<1, 384, 0, stream>>>(p, &scal[1], 1);

    // 6) all 384 reduced solves via Woodbury + deletion identity, loss parts
    mmcl_solve_kernel<<<dim3(BS), 384, 0, stream>>>(X, G, KKd0, Km, p, scal,
                                                    pos_part, neg_part);
    // 7) scalars out
    mmcl_final_kernel<<<1, 384, 0, stream>>>(pos_part, neg_part, out);
}